// PPRPowerIteration_49211735278056
// MI455X (gfx1250) — compile-verified
//
#include <hip/hip_runtime.h>
#include <hip/hip_bf16.h>

#define NN 4096
#define CC 16
#define NITER_ 10
#define KCHUNK 1024   // rows of X staged per LDS buffer (64 KB)

typedef __attribute__((ext_vector_type(2))) float v2f;
typedef __attribute__((ext_vector_type(8))) float v8f;

__device__ __forceinline__ v8f wmma4(v2f a, v2f b, v8f c) {
  // V_WMMA_F32_16X16X4_F32 : D(16x16 f32) = A(16x4 f32) x B(4x16 f32) + C
  return __builtin_amdgcn_wmma_f32_16x16x4_f32(
      /*neg_a=*/false, a, /*neg_b=*/false, b,
      /*c_mod=*/(short)0, c, /*reuse_a=*/false, /*reuse_b=*/false);
}

// Async-copy one K-chunk of X (KCHUNK x 16 f32 = 64 KB) into LDS.
// 256 threads x 16 b128 transfers each; tracked on ASYNCcnt.
__device__ __forceinline__ void async_load_chunk(float* __restrict__ dstLds,
                                                 const float* __restrict__ src,
                                                 int tid) {
#pragma unroll
  for (int i = 0; i < 16; ++i) {
    const int e4 = tid + i * 256;                       // float4 index in chunk
    const unsigned lds_off =
        (unsigned)(uintptr_t)(dstLds + (size_t)e4 * 4); // addr[31:0] == LDS byte offset
    const unsigned long long gaddr =
        (unsigned long long)(uintptr_t)(src + (size_t)e4 * 4);
    asm volatile("global_load_async_to_lds_b128 %0, %1, off"
                 :: "v"(lds_off), "v"(gaddr) : "memory");
  }
}

__device__ __forceinline__ void wait_async_all() {
  asm volatile("s_wait_asynccnt 0x0" ::: "memory");
}

// B-operand fragment from LDS: K=kk+2*lhalf .. +1, N=l15 (ds_load_2addr_b32)
__device__ __forceinline__ v2f lds_b(const float* __restrict__ xb, int kk,
                                     int lhalf, int l15) {
  const int kb = kk + (lhalf << 1);
  v2f b;
  b.x = xb[kb * CC + l15];
  b.y = xb[kb * CC + CC + l15];
  return b;
}

// acc(16x16) = A[row0:row0+16, :] (16 x NN) @ X (NN x 16), f32 WMMA 16x16x4.
// X staged through double-buffered LDS (async copies); two independent WMMA
// chains + 1-iteration software pipeline for memory-level parallelism.
__device__ __forceinline__ v8f gemm_tile(const float* __restrict__ A,
                                         const float* __restrict__ X,
                                         float* __restrict__ Xs,  // [2*KCHUNK*CC]
                                         int row0, int tid) {
  const int lane  = tid & 31;
  const int lhalf = lane >> 4;          // 0: K=+0,+1   1: K=+2,+3
  const int l15   = lane & 15;
  const float* arow = A + (size_t)(row0 + l15) * NN + (lhalf << 1);
  v8f acc0 = {0.f, 0.f, 0.f, 0.f, 0.f, 0.f, 0.f, 0.f};
  v8f acc1 = {0.f, 0.f, 0.f, 0.f, 0.f, 0.f, 0.f, 0.f};

  async_load_chunk(Xs, X, tid);                         // prologue: chunk 0

  for (int c = 0; c < NN / KCHUNK; ++c) {
    wait_async_all();                                   // own wave's copies done
    __syncthreads();                                    // all waves' copies done
    if (c + 1 < NN / KCHUNK)                            // overlap next copy with compute
      async_load_chunk(Xs + ((c + 1) & 1) * (KCHUNK * CC),
                       X + (size_t)(c + 1) * KCHUNK * CC, tid);

    const float* xb = Xs + (c & 1) * (KCHUNK * CC);
    const float* ab = arow + c * KCHUNK;
    if (c + 1 < NN / KCHUNK) {                          // nudge next chunk's A stream
      __builtin_prefetch(ab + KCHUNK, 0, 0);
      __builtin_prefetch(ab + KCHUNK + 512, 0, 0);
    }

    // software pipeline: prologue loads pair 0 (k=0 and k=4)
    v2f a0 = *(const v2f*)(ab);
    v2f a1 = *(const v2f*)(ab + 4);
    v2f b0 = lds_b(xb, 0, lhalf, l15);
    v2f b1 = lds_b(xb, 4, lhalf, l15);
#pragma unroll 4
    for (int kk = 0; kk <= KCHUNK - 16; kk += 8) {      // body: 127 iterations
      v2f na0 = *(const v2f*)(ab + kk + 8);
      v2f na1 = *(const v2f*)(ab + kk + 12);
      v2f nb0 = lds_b(xb, kk + 8, lhalf, l15);
      v2f nb1 = lds_b(xb, kk + 12, lhalf, l15);
      acc0 = wmma4(a0, b0, acc0);
      acc1 = wmma4(a1, b1, acc1);
      a0 = na0; a1 = na1; b0 = nb0; b1 = nb1;
    }
    acc0 = wmma4(a0, b0, acc0);                         // epilogue: pair 1016
    acc1 = wmma4(a1, b1, acc1);

    __syncthreads();                                    // buffer reuse fence
  }
  return acc0 + acc1;                                   // positional layout: exact
}

// H = preds0 - B @ preds0 ;  tmp0 = tau[NITER] * H
__global__ void k_gemm_init(const float* __restrict__ Bm,
                            const float* __restrict__ preds0,
                            const float* __restrict__ tau,
                            float* __restrict__ H, float* __restrict__ tmp0) {
  __shared__ __align__(16) float Xs[2 * KCHUNK * CC];   // 128 KB
  const int tid = threadIdx.x, wave = tid >> 5, lane = tid & 31;
  const int row0 = (blockIdx.x * 8 + wave) * 16;
  v8f acc = gemm_tile(Bm, preds0, Xs, row0, tid);
  const float tN = tau[NITER_];
  const int lhalf = lane >> 4, l15 = lane & 15;
#pragma unroll
  for (int r = 0; r < 8; ++r) {
    const int row = row0 + r + (lhalf << 3);   // C/D: vgpr r -> M=r (lanes0-15), M=r+8 (lanes16-31)
    const int o = row * CC + l15;
    const float h = preds0[o] - acc[r];
    H[o] = h;
    tmp0[o] = tN * h;
  }
}

// tmp_out = tau[it]*H + W @ tmp_in
__global__ void k_gemm_step(const float* __restrict__ Wm,
                            const float* __restrict__ tin,
                            const float* __restrict__ H,
                            const float* __restrict__ tau, int it,
                            float* __restrict__ tout) {
  __shared__ __align__(16) float Xs[2 * KCHUNK * CC];   // 128 KB
  const int tid = threadIdx.x, wave = tid >> 5, lane = tid & 31;
  const int row0 = (blockIdx.x * 8 + wave) * 16;
  v8f acc = gemm_tile(Wm, tin, Xs, row0, tid);
  const float ti = tau[it];
  const int lhalf = lane >> 4, l15 = lane & 15;
#pragma unroll
  for (int r = 0; r < 8; ++r) {
    const int row = row0 + r + (lhalf << 3);
    const int o = row * CC + l15;
    tout[o] = fmaf(ti, H[o], acc[r]);
  }
}

__global__ void k_zero(float4* __restrict__ dst, size_t n4) {
  size_t t = blockIdx.x * (size_t)blockDim.x + threadIdx.x;
  const size_t stride = (size_t)gridDim.x * blockDim.x;
  const float4 z = make_float4(0.f, 0.f, 0.f, 0.f);
  for (size_t i = t; i < n4; i += stride) dst[i] = z;
}

__global__ void k_norms(const float* __restrict__ preds,
                        const float* __restrict__ preds0,
                        float* __restrict__ nrm, float* __restrict__ nrm0) {
  const int j = blockIdx.x * blockDim.x + threadIdx.x;  // 4096
  float s = 0.f, s0 = 0.f;
#pragma unroll
  for (int c = 0; c < CC; ++c) {
    const float a = preds[j * CC + c];  s  = fmaf(a, a, s);
    const float b = preds0[j * CC + c]; s0 = fmaf(b, b, s0);
  }
  nrm[j] = s; nrm0[j] = s0;
}

__global__ void k_gather(const float* __restrict__ preds,
                         const int* __restrict__ idx, float* __restrict__ out) {
  const int t = blockIdx.x * blockDim.x + threadIdx.x;  // 16384
  const int m = t >> 4, c = t & 15;
  out[t] = preds[idx[m] * CC + c];
}

// one wave per row: top-4 smallest of d_w, top-4 largest of d_b, scatter weights
__global__ void k_topk(const float* __restrict__ preds,
                       const float* __restrict__ preds0,
                       const float* __restrict__ nrm,
                       const float* __restrict__ nrm0,
                       float* __restrict__ Wn, float* __restrict__ Bn) {
  const int wave = threadIdx.x >> 5, lane = threadIdx.x & 31;
  const int row = blockIdx.x * 8 + wave;
  float p[CC];
#pragma unroll
  for (int c = 0; c < CC; ++c) p[c] = preds[row * CC + c];
  const float xxi = nrm[row];

  float w0 = INFINITY, w1 = INFINITY, w2 = INFINITY, w3 = INFINITY;
  int   iw0 = 0x7fffffff, iw1 = 0x7fffffff, iw2 = 0x7fffffff, iw3 = 0x7fffffff;
  float b0 = -INFINITY, b1 = -INFINITY, b2 = -INFINITY, b3 = -INFINITY;
  int   ib0 = 0x7fffffff, ib1 = 0x7fffffff, ib2 = 0x7fffffff, ib3 = 0x7fffffff;

  for (int j = lane; j < NN; j += 32) {
    const float* q = preds + j * CC;
    const float* q0 = preds0 + j * CC;
    float dot = 0.f, dot0 = 0.f;
#pragma unroll
    for (int c = 0; c < CC; ++c) {
      dot  = fmaf(p[c], q[c],  dot);
      dot0 = fmaf(p[c], q0[c], dot0);
    }
    const float dw = fmaxf(xxi + nrm[j]  - 2.f * dot,  0.f);
    const float db = fmaxf(xxi + nrm0[j] - 2.f * dot0, 0.f);
    if (dw < w3) {
      w3 = dw; iw3 = j;
      if (w3 < w2) { float tv = w2; w2 = w3; w3 = tv; int ti = iw2; iw2 = iw3; iw3 = ti; }
      if (w2 < w1) { float tv = w1; w1 = w2; w2 = tv; int ti = iw1; iw1 = iw2; iw2 = ti; }
      if (w1 < w0) { float tv = w0; w0 = w1; w1 = tv; int ti = iw0; iw0 = iw1; iw1 = ti; }
    }
    if (db > b3) {
      b3 = db; ib3 = j;
      if (b3 > b2) { float tv = b2; b2 = b3; b3 = tv; int ti = ib2; ib2 = ib3; ib3 = ti; }
      if (b2 > b1) { float tv = b1; b1 = b2; b2 = tv; int ti = ib1; ib1 = ib2; ib2 = ti; }
      if (b1 > b0) { float tv = b0; b0 = b1; b1 = tv; int ti = ib0; ib0 = ib1; ib1 = ti; }
    }
  }

  float rv[4]; int ri[4];
  int ptr = 0;
#pragma unroll
  for (int s = 0; s < 4; ++s) {
    float cv = (ptr == 0) ? w0 : (ptr == 1) ? w1 : (ptr == 2) ? w2 : (ptr == 3) ? w3 : INFINITY;
    int   ci = (ptr == 0) ? iw0 : (ptr == 1) ? iw1 : (ptr == 2) ? iw2 : (ptr == 3) ? iw3 : 0x7fffffff;
    float bv = cv; int bi = ci;
#pragma unroll
    for (int off = 16; off > 0; off >>= 1) {
      const float ov = __shfl_xor(bv, off, 32);
      const int   oi = __shfl_xor(bi, off, 32);
      if (ov < bv || (ov == bv && oi < bi)) { bv = ov; bi = oi; }
    }
    rv[s] = bv; ri[s] = bi;
    if (ptr < 4 && cv == bv && ci == bi) ptr++;
  }

  float sv[4]; int si[4];
  ptr = 0;
#pragma unroll
  for (int s = 0; s < 4; ++s) {
    float cv = (ptr == 0) ? b0 : (ptr == 1) ? b1 : (ptr == 2) ? b2 : (ptr == 3) ? b3 : -INFINITY;
    int   ci = (ptr == 0) ? ib0 : (ptr == 1) ? ib1 : (ptr == 2) ? ib2 : (ptr == 3) ? ib3 : 0x7fffffff;
    float bv = cv; int bi = ci;
#pragma unroll
    for (int off = 16; off > 0; off >>= 1) {
      const float ov = __shfl_xor(bv, off, 32);
      const int   oi = __shfl_xor(bi, off, 32);
      if (ov > bv || (ov == bv && oi < bi)) { bv = ov; bi = oi; }
    }
    sv[s] = bv; si[s] = bi;
    if (ptr < 4 && cv == bv && ci == bi) ptr++;
  }

  if (lane == 0) {
    const float m = rv[3];
    const float den = 3.f * m - (rv[0] + rv[1] + rv[2]);
    float* wr = Wn + (size_t)row * NN;
    wr[ri[0]] = (m - rv[0]) / den;
    wr[ri[1]] = (m - rv[1]) / den;
    wr[ri[2]] = (m - rv[2]) / den;
    const float mb = sv[3];
    const float denb = 3.f * (sv[0] + sv[1] + sv[2]) - mb;
    float* br = Bn + (size_t)row * NN;
    br[si[0]] = (sv[0] - mb) / denb;
    br[si[1]] = (sv[1] - mb) / denb;
    br[si[2]] = (sv[2] - mb) / denb;
  }
}

extern "C" void kernel_launch(void* const* d_in, const int* in_sizes, int n_in,
                              void* d_out, int out_size, void* d_ws, size_t ws_size,
                              hipStream_t stream) {
  const float* preds0 = (const float*)d_in[0];   // 4096x16
  const float* W      = (const float*)d_in[1];   // 4096x4096
  const float* B      = (const float*)d_in[2];   // 4096x4096
  const float* tau    = (const float*)d_in[3];   // 11
  const int*   idx    = (const int*)d_in[4];     // 1024

  float* out = (float*)d_out;                    // 16384 | Wn 4096^2 | Bn 4096^2
  float* Wn  = out + 16384;
  float* Bn  = Wn + (size_t)NN * NN;

  float* ws   = (float*)d_ws;
  float* H    = ws;                 // 65536 f
  float* tA   = ws + 65536;         // 65536 f
  float* tB   = ws + 2 * 65536;     // 65536 f
  float* nrm  = ws + 3 * 65536;     // 4096 f
  float* nrm0 = nrm + NN;           // 4096 f

  const size_t n4 = (2ull * NN * NN) / 4;
  k_zero<<<2048, 256, 0, stream>>>((float4*)Wn, n4);

  k_gemm_init<<<32, 256, 0, stream>>>(B, preds0, tau, H, tA);

  float* tin = tA; float* tout = tB;
  for (int i = NITER_ - 1; i >= 0; --i) {
    k_gemm_step<<<32, 256, 0, stream>>>(W, tin, H, tau, i, tout);
    float* t = tin; tin = tout; tout = t;
  }
  const float* preds = tin;  // final result after 10 steps

  k_norms<<<NN / 256, 256, 0, stream>>>(preds, preds0, nrm, nrm0);
  k_gather<<<16384 / 256, 256, 0, stream>>>(preds, idx, out);
  k_topk<<<NN / 8, 256, 0, stream>>>(preds, preds0, nrm, nrm0, Wn, Bn);
}